// JaxAttention_77936476553270
// MI455X (gfx1250) — compile-verified
//
#include <hip/hip_runtime.h>
#include <math.h>

#define NSEQ 32
#define NH 32
#define KVH 8
#define GQA 4
#define HD 128
#define BPS 64
#define MAXKV 1024
#define ROWF 2048                 // floats per slot: 2*8*128
#define NSLOTS 32768
#define CACHE_F (NSLOTS * ROWF)   // 67108864 floats
#define SCALEF 0.08838834764831845f

typedef float v2f __attribute__((ext_vector_type(2)));
typedef float v4f __attribute__((ext_vector_type(4)));
typedef float v8f __attribute__((ext_vector_type(8)));
typedef int   v4i __attribute__((ext_vector_type(4)));

#if defined(__has_builtin)
# if __has_builtin(__builtin_amdgcn_global_load_async_to_lds_b128)
#  define HAVE_ASYNC_LDS 1
# endif
# if __has_builtin(__builtin_amdgcn_s_wait_asynccnt)
#  define HAVE_WAIT_ASYNC 1
# endif
#endif

#define AS1 __attribute__((address_space(1)))
#define AS3 __attribute__((address_space(3)))
__device__ static inline AS1 v4i* as_global_v4i(const void* p) {
    return (AS1 v4i*)(unsigned long long)p;
}
__device__ static inline AS3 v4i* as_lds_v4i(const void* p) {
    return (AS3 v4i*)(unsigned int)(unsigned long long)p;
}

// ---------------- kernel 1: bulk cache copy (bandwidth bound) ----------------
__global__ __launch_bounds__(256) void copy_cache(const v4f* __restrict__ src,
                                                  v4f* __restrict__ dst, int n4) {
    int i = blockIdx.x * blockDim.x + threadIdx.x;
    int stride = gridDim.x * blockDim.x;
    for (; i < n4; i += stride) {
        v4f val = src[i];                          // temporal: attention re-reads via L2
        __builtin_nontemporal_store(val, dst + i); // output never re-read on device
    }
}

// ---------------- kernel 2: scatter fresh k/v into updated rows -------------
__global__ __launch_bounds__(256) void scatter_kv(const v4f* __restrict__ k,
                                                  const v4f* __restrict__ v,
                                                  const int* __restrict__ slot_mapping,
                                                  v4f* __restrict__ dst) {
    int s = blockIdx.x, t = threadIdx.x;        // 256 threads, 512 f4 per row
    int slot = slot_mapping[s];
    v4f* db = dst + (size_t)slot * (ROWF / 4);
    db[t]       = k[s * 256 + t];               // K half: floats [0,1024)
    db[256 + t] = v[s * 256 + t];               // V half: floats [1024,2048)
}

// ---------------- kernel 3: GQA paged attention with f32 WMMA ----------------
__global__ __launch_bounds__(128) void attn(const float* __restrict__ cache,
                                            const float* __restrict__ q,
                                            const float* __restrict__ knew,
                                            const float* __restrict__ vnew,
                                            const int* __restrict__ block_tables,
                                            const int* __restrict__ seq_lens,
                                            float* __restrict__ out) {
    const int h = blockIdx.x;          // kv head
    const int s = blockIdx.y;          // sequence
    const int w = threadIdx.x >> 5;    // wave id 0..3
    const int lane = threadIdx.x & 31;
    const int n = lane & 15;           // column / M-row index within half-wave
    const int hi = lane >> 4;          // lane-half selects K-offset {0,+2}
    const int seq_len = seq_lens[s];
    const int last = seq_len - 1;

    __shared__ float Qlds[16 * 132];           // rows 4..15 zero-padded
    __shared__ float Klds[4][16 * 132];        // per-wave K tile, padded stride
    __shared__ float Plds[4][16 * 20];         // per-wave P transpose buffer
    __shared__ float red_m[4][4], red_l[4][4];
    __shared__ float red_o[4][4][128];

    // Stage Q (zero-padded to 16 rows) once per block
    for (int i = threadIdx.x; i < 16 * 132; i += 128) Qlds[i] = 0.f;
    __syncthreads();
    for (int i = threadIdx.x; i < GQA * HD; i += 128) {
        int g = i >> 7, d = i & 127;
        Qlds[g * 132 + d] = q[(size_t)s * (NH * HD) + (h * GQA + g) * HD + d];
    }
    __syncthreads();

    float rm[4], rl[4];
    v8f od[8];
#pragma unroll
    for (int v = 0; v < 4; ++v) { rm[v] = -INFINITY; rl[v] = 0.f; }
#pragma unroll
    for (int dt = 0; dt < 8; ++dt) {
#pragma unroll
        for (int j = 0; j < 8; ++j) od[dt][j] = 0.f;
    }

    const int* bt = block_tables + s * BPS;
    float* Kl = Klds[w];
    float* Pl = Plds[w];
    const float* knew_row = knew + (size_t)s * (KVH * HD) + h * HD;
    const float* vnew_row = vnew + (size_t)s * (KVH * HD) + h * HD;

    const int kv_begin = w * 256;
    const int kv_end = (kv_begin + 256 < seq_len) ? kv_begin + 256 : seq_len;

#if HAVE_ASYNC_LDS
    // Issue one tile's 16 rows (512B each, 32 lanes x 16B) on the async pipe.
    auto issue_ktile = [&](int kv0) {
        const int blk = bt[kv0 >> 4];
        const float* kb = cache + (size_t)blk * 16 * ROWF + h * HD;
#pragma unroll
        for (int i = 0; i < 16; ++i) {
            const float* gsrc = (kv0 + i == last)
                                    ? (knew_row + lane * 4)
                                    : (kb + (size_t)i * ROWF + lane * 4);
            __builtin_amdgcn_global_load_async_to_lds_b128(
                as_global_v4i(gsrc), as_lds_v4i(Kl + i * 132 + lane * 4), 0, 0);
        }
    };
    // Prologue: prefetch first tile so the loop always computes behind a fetch.
    if (kv_begin < kv_end) issue_ktile(kv_begin);
#endif

    for (int kv0 = kv_begin; kv0 < kv_end; kv0 += 16) {
        const int blk = bt[kv0 >> 4];                       // one tile == one block
        const float* vbase = cache + (size_t)blk * 16 * ROWF + KVH * HD + h * HD;

#if HAVE_ASYNC_LDS
        // Wait for the tile prefetched during the previous iteration's PV phase.
# if HAVE_WAIT_ASYNC
        __builtin_amdgcn_s_wait_asynccnt(0);
# else
        asm volatile("s_wait_asynccnt 0x0" ::: "memory");
# endif
        asm volatile("" ::: "memory");
#else
        // Fallback: batch 8 rows in distinct VGPRs so 8 loads are in flight
        // before the first waitcnt (avoids load->wait->store serialization).
        const float* kbase = cache + (size_t)blk * 16 * ROWF + h * HD;
#pragma unroll
        for (int half = 0; half < 2; ++half) {
            v4f buf[8];
#pragma unroll
            for (int i = 0; i < 8; ++i) {
                int row = half * 8 + i;
                if (kv0 + row == last) buf[i] = *(const v4f*)(knew_row + lane * 4);
                else buf[i] = *(const v4f*)(kbase + (size_t)row * ROWF + lane * 4);
            }
#pragma unroll
            for (int i = 0; i < 8; ++i)
                *(v4f*)(Kl + (half * 8 + i) * 132 + lane * 4) = buf[i];
        }
#endif

        // --- QK^T: D(16x16) += A(Q 16x4) x B(K^T 4x16), 32 K-steps ---
        v8f sc;
#pragma unroll
        for (int j = 0; j < 8; ++j) sc[j] = 0.f;
#pragma unroll
        for (int c = 0; c < 32; ++c) {
            v2f a = *(const v2f*)(Qlds + n * 132 + 4 * c + 2 * hi);
            v2f b = *(const v2f*)(Kl   + n * 132 + 4 * c + 2 * hi);
            sc = __builtin_amdgcn_wmma_f32_16x16x4_f32(false, a, false, b,
                                                       (short)0, sc, false, false);
        }

#if HAVE_ASYNC_LDS
        // DS ops are in-order per wave: dscnt==0 => all Klds reads above have
        // retired, so the async pipe may overwrite Klds with the NEXT tile
        // while we do softmax + P round-trip + PV (Plds/global only).
        asm volatile("s_wait_dscnt 0x0" ::: "memory");
        if (kv0 + 16 < kv_end) issue_ktile(kv0 + 16);
#endif

        // --- scale + column mask ---
        const bool colvalid = (kv0 + n) < seq_len;
#pragma unroll
        for (int j = 0; j < 8; ++j) sc[j] = colvalid ? sc[j] * SCALEF : -INFINITY;

        // --- online softmax on valid rows 0..3 (VGPRs 0..3, lanes 0..15) ---
#pragma unroll
        for (int v = 0; v < 4; ++v) {
            float x = sc[v];
            x = fmaxf(x, __shfl_xor(x, 8, 32));
            x = fmaxf(x, __shfl_xor(x, 4, 32));
            x = fmaxf(x, __shfl_xor(x, 2, 32));
            x = fmaxf(x, __shfl_xor(x, 1, 32));
            float nm = fmaxf(rm[v], x);
            float r = __expf(rm[v] - nm);
            rm[v] = nm;
            float p = __expf(sc[v] - nm);      // -inf cols -> 0
            sc[v] = p;
            float ps = p;
            ps += __shfl_xor(ps, 8, 32);
            ps += __shfl_xor(ps, 4, 32);
            ps += __shfl_xor(ps, 2, 32);
            ps += __shfl_xor(ps, 1, 32);
            rl[v] = rl[v] * r + ps;
#pragma unroll
            for (int dt = 0; dt < 8; ++dt) od[dt][v] *= r;   // rescale row v
        }

        // --- D-layout -> A-layout via LDS; zero all rows except 0..3 ---
#pragma unroll
        for (int v = 0; v < 8; ++v) {
            float pv = (hi == 0 && v < 4) ? sc[v] : 0.f;
            Pl[(v + 8 * hi) * 20 + n] = pv;
        }

        // --- PV: out(16x128) += P(16x16) x V(16x128), tiled 8 x (4 K-steps) ---
#pragma unroll
        for (int dt = 0; dt < 8; ++dt) {
#pragma unroll
            for (int c = 0; c < 4; ++c) {
                int k0 = 4 * c + 2 * hi;
                v2f a = *(const v2f*)(Pl + n * 20 + 4 * c + 2 * hi);
                int kk0 = kv0 + k0;
                float bx = (kk0 == last)     ? vnew_row[dt * 16 + n]
                                             : vbase[(size_t)k0 * ROWF + dt * 16 + n];
                float by = (kk0 + 1 == last) ? vnew_row[dt * 16 + n]
                                             : vbase[(size_t)(k0 + 1) * ROWF + dt * 16 + n];
                v2f b; b.x = bx; b.y = by;
                od[dt] = __builtin_amdgcn_wmma_f32_16x16x4_f32(false, a, false, b,
                                                               (short)0, od[dt], false, false);
            }
        }
    }

    // --- publish per-wave partials ---
    if (lane == 0) {
#pragma unroll
        for (int v = 0; v < 4; ++v) { red_m[w][v] = rm[v]; red_l[w][v] = rl[v]; }
    }
    if (hi == 0) {
#pragma unroll
        for (int dt = 0; dt < 8; ++dt)
#pragma unroll
            for (int v = 0; v < 4; ++v)
                red_o[w][v][dt * 16 + n] = od[dt][v];
    }
    __syncthreads();

    // --- cross-wave flash combine; 128 threads = 128 head-dims ---
    const int d = threadIdx.x;
#pragma unroll
    for (int v = 0; v < 4; ++v) {
        float gm = -INFINITY;
#pragma unroll
        for (int ww = 0; ww < 4; ++ww) gm = fmaxf(gm, red_m[ww][v]);
        float L = 0.f, acc = 0.f;
#pragma unroll
        for (int ww = 0; ww < 4; ++ww) {
            float f = __expf(red_m[ww][v] - gm);
            L   += f * red_l[ww][v];
            acc += f * red_o[ww][v][d];
        }
        out[(size_t)s * (NH * HD) + (h * GQA + v) * HD + d] = acc / L;
    }
}

extern "C" void kernel_launch(void* const* d_in, const int* in_sizes, int n_in,
                              void* d_out, int out_size, void* d_ws, size_t ws_size,
                              hipStream_t stream) {
    const float* cache        = (const float*)d_in[0];
    const float* q            = (const float*)d_in[1];
    const float* k            = (const float*)d_in[2];
    const float* v            = (const float*)d_in[3];
    const int*   slot_mapping = (const int*)d_in[4];
    const int*   block_tables = (const int*)d_in[5];
    const int*   seq_lens     = (const int*)d_in[6];
    float* out = (float*)d_out;

    // Output 1: new_cache = copy + scatter of fresh rows
    copy_cache<<<8192, 256, 0, stream>>>((const v4f*)cache, (v4f*)out, CACHE_F / 4);
    scatter_kv<<<NSEQ, 256, 0, stream>>>((const v4f*)k, (const v4f*)v, slot_mapping, (v4f*)out);

    // Output 2: attention (reads original cache + fresh k/v; independent of copy)
    dim3 grid(KVH, NSEQ);
    attn<<<grid, 128, 0, stream>>>(cache, q, k, v, block_tables, seq_lens,
                                   out + (size_t)CACHE_F);
    (void)in_sizes; (void)n_in; (void)out_size; (void)d_ws; (void)ws_size;
}